// DirectVoxGO_70128226009813
// MI455X (gfx1250) — compile-verified
//
#include <hip/hip_runtime.h>

// DirectVoxGO volume render, MI455X (gfx1250, wave32).
// One wave per ray. Wave32 ds_bpermute segmented scan; xyz stream double-
// buffered into LDS with GLOBAL_LOAD_ASYNC_TO_LDS_B32 (ASYNCcnt); density
// cachelines prefetched one chunk ahead with global_prefetch_b8.

#define GD    256        // density grid dim
#define GSX   65536      // x stride in density grid (G*G)
#define GSY   256        // y stride
#define PCS   65536      // plane channel stride (256*256)
#define WPB   8          // waves per block

typedef __attribute__((address_space(1))) int gint;   // global int (AS1)
typedef __attribute__((address_space(3))) int lint;   // LDS int    (AS3)

__device__ __forceinline__ float fexp(float x) {        // e^x via v_exp_f32
  return __builtin_amdgcn_exp2f(x * 1.44269504088896340736f);
}
__device__ __forceinline__ float flog2(float x) { return __builtin_amdgcn_logf(x); }
__device__ __forceinline__ float frcp(float x)  { return __builtin_amdgcn_rcpf(x); }

// ln(1+u) for u in (0,1]: polynomial for tiny u (1+u would round away the
// info), native v_log otherwise.
__device__ __forceinline__ float log1p_pos(float u) {
  float small = u * (1.f + u * (-0.5f + u * (0.33333333f + u * (-0.25f))));
  float big   = flog2(1.f + u) * 0.69314718055994530942f;
  return (u < 0.0078125f) ? small : big;
}

// softplus(x) = max(x,0) + log1p(exp(-|x|)); accurate for x ~ -13.8
__device__ __forceinline__ float softplus_acc(float x) {
  return fmaxf(x, 0.f) + log1p_pos(fexp(-fabsf(x)));
}

// 1 - e^t for t <= 0 == -expm1(t); polynomial near 0 avoids cancellation.
__device__ __forceinline__ float one_minus_exp(float t) {
  float poly  = -t * (1.f + t * (0.5f + t * (0.16666667f + t * 0.041666667f)));
  float exact = 1.f - fexp(t);
  return (t > -0.0625f) ? poly : exact;
}

// continuous coord -> (cell index, fraction), matching the reference clamps
__device__ __forceinline__ void coord(float v, int& i0, float& f) {
  float p = fminf(fmaxf((v + 1.f) * 127.5f, 0.f), 255.f);
  int i = (int)p;            // floor (p >= 0)
  if (i > GD - 2) i = GD - 2;
  f = p - (float)i;
  i0 = i;
}

// lane permute (wave32): dst[lane] = src[idx]
__device__ __forceinline__ float bperm_f(int srcLane, float v) {
  return __int_as_float(__builtin_amdgcn_ds_bpermute(srcLane << 2, __float_as_int(v)));
}

// per-lane async copy of 12 bytes (one xyz sample) global -> LDS
__device__ __forceinline__ void async_xyz(const float* gp, float* lp) {
  gint* g = (gint*)gp;
  lint* l = (lint*)lp;
  __builtin_amdgcn_global_load_async_to_lds_b32(g, l, 0, 0);
  __builtin_amdgcn_global_load_async_to_lds_b32(g, l, 4, 0);
  __builtin_amdgcn_global_load_async_to_lds_b32(g, l, 8, 0);
}

// 3-channel bilinear accumulate from a [3,256,256] plane
__device__ __forceinline__ void bil3(const float* __restrict__ plane,
                                     int i0, int j0, float fi, float fj,
                                     float f[3]) {
  const int   b   = i0 * 256 + j0;
  const float w00 = (1.f - fi) * (1.f - fj);
  const float w01 = (1.f - fi) * fj;
  const float w10 = fi * (1.f - fj);
  const float w11 = fi * fj;
#pragma unroll
  for (int c = 0; c < 3; ++c) {
    const float* p = plane + c * PCS + b;
    f[c] += p[0] * w00 + p[1] * w01 + p[256] * w10 + p[257] * w11;
  }
}

__global__ __launch_bounds__(256) void dvgo_render_kernel(
    const float*     __restrict__ xyz,      // [M,3]
    const long long* __restrict__ ray_id,   // [M] sorted int64
    const float*     __restrict__ grid,     // [256,256,256]
    const float*     __restrict__ k0_xy,    // [3,256,256]
    const float*     __restrict__ k0_yz,
    const float*     __restrict__ k0_zx,
    float*           __restrict__ out,      // [N,3]
    int M, int NR)
{
  __shared__ float xbuf[2][WPB][32 * 3];    // double-buffered xyz staging

  const int lane  = threadIdx.x & 31;
  const int wslot = threadIdx.x >> 5;
  const int ray   = blockIdx.x * (blockDim.x >> 5) + wslot;
  if (ray >= NR) return;   // uniform per wave

  const float ACT_SHIFT = -13.815509557964774f;  // log(1/(1-1e-6) - 1)

  // --- segment bounds via binary search in sorted int64 ray_id (uniform) ---
  int lo, hi;
  {
    long long key = (long long)ray;
    int a = 0, b = M;
    while (a < b) { int mid = (a + b) >> 1; if (ray_id[mid] < key) a = mid + 1; else b = mid; }
    lo = a;
    b = M;
    long long key2 = key + 1;
    while (a < b) { int mid = (a + b) >> 1; if (ray_id[mid] < key2) a = mid + 1; else b = mid; }
    hi = a;
  }

  float running = 0.f;                    // within-ray cumulative sum of logs
  float acc0 = 0.f, acc1 = 0.f, acc2 = 0.f;

  if (lo < hi) {
    // chunk 0 xyz straight into registers
    int s0 = lo + lane; if (s0 > hi - 1) s0 = hi - 1;
    const float* xp = xyz + 3 * (size_t)s0;
    float X = xp[0], Y = xp[1], Z = xp[2];

    // launch async stage of chunk 1 into LDS buffer 0
    int buf = 0;
    if (lo + 32 < hi) {
      int sn = lo + 32 + lane; if (sn > hi - 1) sn = hi - 1;
      async_xyz(xyz + 3 * (size_t)sn, &xbuf[0][wslot][lane * 3]);
    }

    for (int base = lo; base < hi; base += 32) {
      const bool valid = (base + lane) < hi;

      int xi, yi, zi; float fx, fy, fz;
      coord(X, xi, fx); coord(Y, yi, fy); coord(Z, zi, fz);

      // ---- trilinear density (gathers; 64MB grid lives in 192MB L2) ----
      const float* g = grid + (xi * GSX + yi * GSY + zi);
      const float c000 = g[0],           c001 = g[1];
      const float c010 = g[GSY],         c011 = g[GSY + 1];
      const float c100 = g[GSX],         c101 = g[GSX + 1];
      const float c110 = g[GSX + GSY],   c111 = g[GSX + GSY + 1];
      float c00 = c000 + (c100 - c000) * fx;
      float c10 = c010 + (c110 - c010) * fx;
      float c01 = c001 + (c101 - c001) * fx;
      float c11 = c011 + (c111 - c011) * fx;
      float cA  = c00 + (c10 - c00) * fy;
      float cB  = c01 + (c11 - c01) * fy;
      float den = cA + (cB - cA) * fz;

      float logs = -0.5f * softplus_acc(den + ACT_SHIFT);
      if (!valid) logs = 0.f;                     // inert in the scan
      const float alpha = one_minus_exp(logs);    // -expm1(logs), accurate

      // ---- wave32 inclusive prefix-sum of logs via ds_bpermute ----
      float incl = logs;
#pragma unroll
      for (int d = 1; d < 32; d <<= 1) {
        float t = bperm_f(lane - d, incl);
        if (lane >= d) incl += t;
      }
      const float excl = incl - logs;
      const float T = fexp(running + excl);
      const float w = valid ? (T * alpha) : 0.f;

      // ---- tri-plane color (3x bilinear, planes are L2/L0 resident) ----
      float feat[3] = {0.f, 0.f, 0.f};
      bil3(k0_xy, xi, yi, fx, fy, feat);
      bil3(k0_yz, yi, zi, fy, fz, feat);
      bil3(k0_zx, zi, xi, fz, fx, feat);
      acc0 += w * frcp(1.f + fexp(-feat[0]));     // sigmoid
      acc1 += w * frcp(1.f + fexp(-feat[1]));
      acc2 += w * frcp(1.f + fexp(-feat[2]));

      // carry this chunk's total (lane 31 inclusive sum) -> uniform
      running += __int_as_float(__builtin_amdgcn_readlane(__float_as_int(incl), 31));

      // ---- retire async stage: pull chunk c+1 from LDS, stage c+2 ----
      const int nbase = base + 32;
      if (nbase < hi) {
        __builtin_amdgcn_s_wait_asynccnt(0);      // chunk c+1 now in LDS
        const float* lp = &xbuf[buf][wslot][lane * 3];
        X = lp[0]; Y = lp[1]; Z = lp[2];
        const int n2 = nbase + 32;
        if (n2 < hi) {                            // stage chunk c+2
          int s2 = n2 + lane; if (s2 > hi - 1) s2 = hi - 1;
          async_xyz(xyz + 3 * (size_t)s2, &xbuf[buf ^ 1][wslot][lane * 3]);
        }
        buf ^= 1;
        // prefetch chunk c+1's density cachelines (global_prefetch_b8)
        int pxi, pyi, pzi; float pf;
        coord(X, pxi, pf); coord(Y, pyi, pf); coord(Z, pzi, pf);
        const float* gp = grid + (pxi * GSX + pyi * GSY + pzi);
        __builtin_prefetch(gp,             0, 1);
        __builtin_prefetch(gp + GSY,       0, 1);
        __builtin_prefetch(gp + GSX,       0, 1);
        __builtin_prefetch(gp + GSX + GSY, 0, 1);
      }
    }
  }

  // ---- cross-lane butterfly reduction of the 3 accumulators ----
#pragma unroll
  for (int d = 16; d >= 1; d >>= 1) {
    acc0 += bperm_f(lane ^ d, acc0);
    acc1 += bperm_f(lane ^ d, acc1);
    acc2 += bperm_f(lane ^ d, acc2);
  }

  if (lane == 0) {
    const float ainv = fexp(running);             // alphainv_last
    float* o = out + 3 * (size_t)ray;
    o[0] = acc0 + ainv;
    o[1] = acc1 + ainv;
    o[2] = acc2 + ainv;
  }
}

extern "C" void kernel_launch(void* const* d_in, const int* in_sizes, int n_in,
                              void* d_out, int out_size, void* d_ws, size_t ws_size,
                              hipStream_t stream) {
  const float*     xyz    = (const float*)d_in[0];
  const long long* ray_id = (const long long*)d_in[1];   // int64
  const float*     grid   = (const float*)d_in[2];
  const float*     k0_xy  = (const float*)d_in[3];
  const float*     k0_yz  = (const float*)d_in[4];
  const float*     k0_zx  = (const float*)d_in[5];
  // d_in[6] is N (device scalar); derive host-side instead:
  float* out = (float*)d_out;
  const int M  = in_sizes[0] / 3;
  const int NR = out_size / 3;

  const int blocks = (NR + WPB - 1) / WPB;
  dvgo_render_kernel<<<blocks, WPB * 32, 0, stream>>>(
      xyz, ray_id, grid, k0_xy, k0_yz, k0_zx, out, M, NR);
}